// ReachabilityPolicy_71184787964323
// MI455X (gfx1250) — compile-verified
//
#include <hip/hip_runtime.h>
#include <stdint.h>

#define NOVELTY_THRESHOLD 0.5f

// ---------------------------------------------------------------------------
// Kernel A: per-env scalar decisions (1024 envs -> trivial).
//   out_mask[e] = mem_mask_new[e] as float (harness output buffer is float)
//   ws_flag[e]  = slot to overwrite, or -1 if this env does not write
// ---------------------------------------------------------------------------
__global__ void decide_kernel(const float* __restrict__ sim,
                              const int*   __restrict__ mem_mask,
                              const int*   __restrict__ rand_idx,
                              float* __restrict__ out_mask,
                              int*   __restrict__ ws_flag,
                              int E, int M) {
    int e = blockIdx.x * blockDim.x + threadIdx.x;
    if (e >= E) return;
    const bool add  = sim[e] > NOVELTY_THRESHOLD;
    const int  mm   = mem_mask[e];
    const bool full = (mm == M);
    const int  newmask = mm + (((!full) && add) ? 1 : 0);
    int aidx = full ? rand_idx[e] : (newmask - 1 > 0 ? newmask - 1 : 0);
    out_mask[e] = (float)newmask;
    ws_flag[e]  = add ? aidx : -1;
}

// ---------------------------------------------------------------------------
// Kernel B: bulk 400 MB streaming copy memory -> d_out through the gfx1250
// async global<->LDS DMA path (ASYNCcnt). Each thread moves 4 x float4:
//   global_load_async_to_lds_b128  th:TH_LOAD_NT   (memory -> LDS)
//   s_wait_asynccnt 0
//   global_store_async_from_lds_b128 th:TH_STORE_NT (LDS -> d_out)
// Non-temporal hints keep this 2x400MB single-use stream from evicting the
// 192MB L2. A wave only reads back LDS addresses it wrote itself, so the
// per-wave s_wait_asynccnt is sufficient ordering: no workgroup barrier.
// Block tile = 256 threads * 64 B = 16 KB of LDS.
// ---------------------------------------------------------------------------
__global__ void bulk_copy_kernel(const float* __restrict__ src,
                                 float* __restrict__ dst,
                                 unsigned long long n4 /* total float4 count */) {
    __shared__ float lds_buf[4096]; // 16 KB
    const unsigned t = threadIdx.x;
    const unsigned long long base4 = (unsigned long long)blockIdx.x * 1024ull;

    // Flat address of an LDS object has the byte offset in addr[31:0]
    // (aperture layout), which is what the async ops' LDS-address VGPR
    // expects (a byte offset into the workgroup's LDS).
    const unsigned lds_base = (unsigned)(uintptr_t)(&lds_buf[0]);

#pragma unroll
    for (int k = 0; k < 4; ++k) {
        unsigned long long e4 = base4 + (unsigned)(k * 256) + t;
        if (e4 < n4) {
            unsigned long long gaddr = (unsigned long long)(uintptr_t)(src) + e4 * 16ull;
            unsigned laddr = lds_base + (unsigned)(k * 256 + t) * 16u;
            asm volatile("global_load_async_to_lds_b128 %0, %1, off th:TH_LOAD_NT"
                         :
                         : "v"(laddr), "v"(gaddr)
                         : "memory");
        }
    }

    asm volatile("s_wait_asynccnt 0x0" ::: "memory");

#pragma unroll
    for (int k = 0; k < 4; ++k) {
        unsigned long long e4 = base4 + (unsigned)(k * 256) + t;
        if (e4 < n4) {
            unsigned long long gaddr = (unsigned long long)(uintptr_t)(dst) + e4 * 16ull;
            unsigned laddr = lds_base + (unsigned)(k * 256 + t) * 16u;
            asm volatile("global_store_async_from_lds_b128 %0, %1, off th:TH_STORE_NT"
                         :
                         : "v"(gaddr), "v"(laddr)
                         : "memory");
        }
    }
    // Async stores drain at s_endpgm (implicit wait-idle).
}

// ---------------------------------------------------------------------------
// Kernel C: per-env row replacement. One block per env; 128 threads move the
// 512-float row as float4. Envs with flag < 0 exit immediately.
// ---------------------------------------------------------------------------
__global__ void replace_row_kernel(const float* __restrict__ obs,
                                   const int*   __restrict__ ws_flag,
                                   float* __restrict__ dst,
                                   int M, int D) {
    const int e = blockIdx.x;
    const int aidx = ws_flag[e];
    if (aidx < 0) return;
    const float4* s = (const float4*)(obs + (size_t)e * D);
    float4*       d = (float4*)(dst + ((size_t)e * M + (size_t)aidx) * D);
    for (int i = threadIdx.x; i < (D >> 2); i += blockDim.x) {
        d[i] = s[i];
    }
}

// ---------------------------------------------------------------------------
// Inputs (setup_inputs order):
//   d_in[0] memory       [E, M, D] f32
//   d_in[1] obs_features [E, D]    f32
//   d_in[2] similarities [E]       f32
//   d_in[3] mem_mask     [E]       i32
//   d_in[4] rand_idx     [E]       i32
// Output: memory_new (E*M*D f32) ++ mem_mask_new (E, written as float)
// ---------------------------------------------------------------------------
extern "C" void kernel_launch(void* const* d_in, const int* in_sizes, int n_in,
                              void* d_out, int out_size, void* d_ws, size_t ws_size,
                              hipStream_t stream) {
    const float* memory = (const float*)d_in[0];
    const float* obs    = (const float*)d_in[1];
    const float* sim    = (const float*)d_in[2];
    const int*   mmask  = (const int*)d_in[3];
    const int*   ridx   = (const int*)d_in[4];

    const int E = in_sizes[2];                 // 1024
    const int D = in_sizes[1] / E;             // 512
    const long long total = (long long)in_sizes[0];
    const int M = (int)(total / ((long long)E * D)); // 200

    float* out_mem  = (float*)d_out;
    float* out_mask = out_mem + (size_t)total;
    int*   ws_flag  = (int*)d_ws;              // E ints of scratch

    // A: decisions
    {
        dim3 grid((E + 255) / 256), block(256);
        decide_kernel<<<grid, block, 0, stream>>>(sim, mmask, ridx,
                                                  out_mask, ws_flag, E, M);
    }

    // B: bulk copy via async DMA. Each block moves 1024 float4 (16 KB).
    {
        const unsigned long long n4 = (unsigned long long)(total / 4);
        const unsigned long long nblocks = (n4 + 1023ull) / 1024ull;
        dim3 grid((unsigned)nblocks), block(256);
        bulk_copy_kernel<<<grid, block, 0, stream>>>(memory, out_mem, n4);
    }

    // C: overwrite replaced rows
    {
        dim3 grid(E), block(128);
        replace_row_kernel<<<grid, block, 0, stream>>>(obs, ws_flag, out_mem, M, D);
    }
}